// CBOWNegSampling_89103391523056
// MI455X (gfx1250) — compile-verified
//
#include <hip/hip_runtime.h>

typedef __attribute__((ext_vector_type(2))) float v2f;
typedef __attribute__((ext_vector_type(8))) float v8f;

#define DIM    50
#define KPAD   52           // DIM padded to 13 chunks of K=4 (pad dims 50,51 = 0 in A)
#define NCHUNK 13
#define CTXW   10
#define NEGK   10
#define NT     (1 + NEGK)
#define TPB    8            // tiles (waves) per block
#define SPT    16           // samples per tile

__global__ void cbow_zero_kernel(float* __restrict__ out) {
    out[0] = 0.0f;
}

__global__ __launch_bounds__(256) void cbow_wmma_kernel(
    const int*   __restrict__ ctxIdx,   // [B, 10]
    const int*   __restrict__ posT,     // [B]
    const int*   __restrict__ negS,     // [B, 10]
    const float* __restrict__ inW,      // [VOCAB, 50]
    const float* __restrict__ outW,     // [VOCAB, 50]
    float*       __restrict__ out,      // scalar accumulator
    int Btot)
{
    __shared__ float ctxLds[TPB][SPT * KPAD];
    __shared__ float blockSum;

    const int lane = threadIdx.x & 31;
    const int wv   = threadIdx.x >> 5;
    const int s0   = (blockIdx.x * TPB + wv) * SPT;

    if (threadIdx.x == 0) blockSum = 0.0f;

    // ---------- Phase 1: context means -> LDS ----------
    // float2 per lane: lanes 0..24 cover dims 0..49; lane 25 writes the zero pad (50,51)
    const float inv = 1.0f / (float)CTXW;
    for (int m = 0; m < SPT; ++m) {
        const int* ib = ctxIdx + (size_t)(s0 + m) * CTXW;
        v2f c = {0.0f, 0.0f};
        if (lane < 25) {
            #pragma unroll
            for (int j = 0; j < CTXW; ++j) {
                const float* row = inW + (size_t)ib[j] * DIM;
                v2f r = *(const v2f*)(row + 2 * lane);     // global_load_b64, coalesced
                c.x += r.x; c.y += r.y;
            }
        }
        if (lane < 26) {
            v2f o;
            o.x = (lane < 25) ? c.x * inv : 0.0f;
            o.y = (lane < 25) ? c.y * inv : 0.0f;
            *(v2f*)(&ctxLds[wv][m * KPAD + 2 * lane]) = o; // ds_store_b64
        }
    }
    __syncthreads();   // uniform across block (all tiles full: B % 128 == 0)

    // ---------- Phase 2: 11 batched 16-dot-products via V_WMMA_F32_16X16X4_F32 ----------
    // A (16x4 f32): M = lane%16, per-lane K = {koff, koff+1}, koff = lane<16 ? 0 : 2
    // B (4x16 f32): N = lane%16, same K striping -> symmetric float2 loads
    const int n    = lane & 15;
    const int koff = (lane < 16) ? 0 : 2;
    const float* myrow = &ctxLds[wv][n * KPAD];

    v2f a[NCHUNK];
    #pragma unroll
    for (int c = 0; c < NCHUNK; ++c)
        a[c] = *(const v2f*)(myrow + c * 4 + koff);        // ds_load_b64

    // One-hot diagonal extractor, computed once (replaces per-t cmp/cndmask chains):
    //   sample i in [0,8)  lives on lane i      in acc[i]
    //   sample i in [8,16) lives on lane i+16   in acc[i-8]
    const int sel = (lane < 8) ? lane : ((lane >= 24) ? (lane - 24) : -1);
    const float ownerf = (sel >= 0) ? 1.0f : 0.0f;
    float oneh[8];
    #pragma unroll
    for (int i = 0; i < 8; ++i) oneh[i] = (sel == i) ? 1.0f : 0.0f;

    const int bsmp = s0 + n;

    // Prefetch all 11 target rows early (global_prefetch_b8) to hide L2 gather latency.
    #pragma unroll
    for (int t = 0; t < NT; ++t) {
        const int tg = (t == 0) ? posT[bsmp] : negS[(size_t)bsmp * NEGK + (t - 1)];
        __builtin_prefetch(outW + (size_t)tg * DIM, 0, 1);
    }

    float lossAcc = 0.0f;
    #pragma unroll 1
    for (int t = 0; t < NT; ++t) {
        const int tg = (t == 0) ? posT[bsmp] : negS[(size_t)bsmp * NEGK + (t - 1)];
        const float* erow = outW + (size_t)tg * DIM;

        v8f acc = {0.f, 0.f, 0.f, 0.f, 0.f, 0.f, 0.f, 0.f};
        #pragma unroll
        for (int c = 0; c < NCHUNK; ++c) {
            // Chunk 12 feeds K=48..51; A's K=50,51 are zero, so both half-waves can
            // load dims {48,49} -> uniform offset 48, branch-free and in-bounds.
            const int off = (c == NCHUNK - 1) ? (DIM - 2) : (c * 4 + koff);
            const v2f bv = *(const v2f*)(erow + off);      // global_load_b64, 8B aligned
            acc = __builtin_amdgcn_wmma_f32_16x16x4_f32(
                      false, a[c], false, bv, (short)0, acc, false, false);
        }

        // Branch-free diagonal extraction: 8 FMAs against the precomputed one-hot.
        float sv = 0.0f;
        #pragma unroll
        for (int i = 0; i < 8; ++i) sv += oneh[i] * acc[i];

        const float x   = (t == 0) ? sv : -sv;             // pos: +score, neg: -score
        const float sig = 1.0f / (1.0f + __expf(-x));
        lossAcc += ownerf * __logf(sig + 1e-10f);
    }

    // ---------- Reduction: wave -> LDS (ds_add_f32) -> one global atomic per block ----------
    float v = lossAcc;                                     // nonzero on 16 owner lanes
    #pragma unroll
    for (int off = 16; off > 0; off >>= 1)
        v += __shfl_down(v, off, 32);
    if (lane == 0)
        atomicAdd(&blockSum, v);
    __syncthreads();
    if (threadIdx.x == 0)
        atomicAdd(out, blockSum * (-1.0f / (float)Btot));  // global_atomic_add_f32
}

extern "C" void kernel_launch(void* const* d_in, const int* in_sizes, int n_in,
                              void* d_out, int out_size, void* d_ws, size_t ws_size,
                              hipStream_t stream) {
    const int*   ctx  = (const int*)d_in[0];    // context_idxs [B,10]
    const int*   pos  = (const int*)d_in[1];    // pos_target  [B]
    const int*   neg  = (const int*)d_in[2];    // neg_samples [B,10]
    const float* inW  = (const float*)d_in[3];  // in_embed_W
    const float* outW = (const float*)d_in[4];  // out_embed_W
    float*       out  = (float*)d_out;

    const int Btot = in_sizes[1];               // B = 131072
    const int blocks = Btot / (TPB * SPT);      // 131072 / 128 = 1024

    cbow_zero_kernel<<<1, 1, 0, stream>>>(out);
    cbow_wmma_kernel<<<blocks, 256, 0, stream>>>(ctx, pos, neg, inW, outW, out, Btot);
}